// MipRayMarcher2_10187662426462
// MI455X (gfx1250) — compile-verified
//
#include <hip/hip_runtime.h>
#include <stdint.h>

#define NRAYS      65536
#define SAMPLES    96
#define NMID       95
#define NW_PER_BLK 4
#define THREADS    (NW_PER_BLK * 32)
#define BLOCKS     2048
#define RAY_BYTES  3072
#define COL_OFF    0
#define CRD_OFF    1152
#define DEN_OFF    2304
#define DEP_OFF    2688

// Output layout (flat concat, reference return order)
#define OFF_RGB    0
#define OFF_DEPTH  196608
#define OFF_W      262144
#define OFF_PNT    6488064
#define OFF_TAU    6684672

// CDNA5 async global->LDS copy (GV mode, 64-bit vaddr, same imm offset on both sides)
#define ASYNC_B32(ldsa, ga, off)                                            \
  asm volatile("global_load_async_to_lds_b32 %0, %1, off offset:" #off      \
               :: "v"(ldsa), "v"(ga) : "memory")

__device__ __forceinline__ unsigned encf(float x) {
  unsigned b = __float_as_uint(x);
  return (b & 0x80000000u) ? ~b : (b | 0x80000000u);
}
__device__ __forceinline__ float decf(unsigned e) {
  unsigned b = (e & 0x80000000u) ? (e & 0x7FFFFFFFu) : ~e;
  return __uint_as_float(b);
}

__device__ __forceinline__ float wave_sum(float v) {
#pragma unroll
  for (int d = 16; d >= 1; d >>= 1) v += __shfl_xor(v, d, 32);
  return v;
}

// Stage one ray (colors 288f, coords 288f, dens 96f, depths 96f) into LDS via 24 async b32 ops.
__device__ __forceinline__ void stage_ray(const float* colors, const float* coords,
                                          const float* dens, const float* deps,
                                          int ray, uint32_t lbase, int lane) {
  uint64_t gc = (uint64_t)(uintptr_t)(colors + (size_t)ray * 288) + (uint32_t)(lane * 4);
  uint64_t gx = (uint64_t)(uintptr_t)(coords + (size_t)ray * 288) + (uint32_t)(lane * 4);
  uint64_t gd = (uint64_t)(uintptr_t)(dens   + (size_t)ray * 96)  + (uint32_t)(lane * 4);
  uint64_t gz = (uint64_t)(uintptr_t)(deps   + (size_t)ray * 96)  + (uint32_t)(lane * 4);
  uint32_t lc = lbase + COL_OFF + lane * 4;
  uint32_t lx = lbase + CRD_OFF + lane * 4;
  uint32_t ld = lbase + DEN_OFF + lane * 4;
  uint32_t lz = lbase + DEP_OFF + lane * 4;
  ASYNC_B32(lc, gc, 0);   ASYNC_B32(lc, gc, 128); ASYNC_B32(lc, gc, 256);
  ASYNC_B32(lc, gc, 384); ASYNC_B32(lc, gc, 512); ASYNC_B32(lc, gc, 640);
  ASYNC_B32(lc, gc, 768); ASYNC_B32(lc, gc, 896); ASYNC_B32(lc, gc, 1024);
  ASYNC_B32(lx, gx, 0);   ASYNC_B32(lx, gx, 128); ASYNC_B32(lx, gx, 256);
  ASYNC_B32(lx, gx, 384); ASYNC_B32(lx, gx, 512); ASYNC_B32(lx, gx, 640);
  ASYNC_B32(lx, gx, 768); ASYNC_B32(lx, gx, 896); ASYNC_B32(lx, gx, 1024);
  ASYNC_B32(ld, gd, 0);   ASYNC_B32(ld, gd, 128); ASYNC_B32(ld, gd, 256);
  ASYNC_B32(lz, gz, 0);   ASYNC_B32(lz, gz, 128); ASYNC_B32(lz, gz, 256);
}

__global__ void rm_init_ws(unsigned* ws) {
  ws[0] = 0xFFFFFFFFu;  // running min (encoded)
  ws[1] = 0x00000000u;  // running max (encoded)
}

// Post-pass: clamp composite_depth to the global depth min/max found by rm_main.
__global__ void rm_clamp(float* __restrict__ outD, const unsigned* __restrict__ ws) {
  int i = blockIdx.x * blockDim.x + threadIdx.x;  // launched exactly NRAYS threads
  float gmin = decf(ws[0]);
  float gmax = decf(ws[1]);
  float d = outD[i];
  outD[i] = fminf(fmaxf(d, gmin), gmax);
}

__global__ __launch_bounds__(THREADS)
void rm_main(const float* __restrict__ colors, const float* __restrict__ dens,
             const float* __restrict__ deps, const float* __restrict__ coords,
             const int* __restrict__ wbp, float* __restrict__ out,
             unsigned* __restrict__ ws) {
  __shared__ unsigned char smem[NW_PER_BLK * 2 * RAY_BYTES];
  const int lane = threadIdx.x & 31;
  const int wib  = threadIdx.x >> 5;
  const int wid  = blockIdx.x * NW_PER_BLK + wib;
  const int stride = gridDim.x * NW_PER_BLK;

  unsigned char* buf0 = smem + (size_t)(wib * 2 + 0) * RAY_BYTES;
  unsigned char* buf1 = smem + (size_t)(wib * 2 + 1) * RAY_BYTES;
  const uint32_t l0 = (uint32_t)(uintptr_t)buf0;  // LDS byte offset (low 32 bits of flat ptr)
  const uint32_t l1 = (uint32_t)(uintptr_t)buf1;

  const int wb = wbp[0];

  float* outRGB = out + OFF_RGB;
  float* outD   = out + OFF_DEPTH;
  float* outW   = out + OFF_W;
  float* outP   = out + OFF_PNT;
  float* outT   = out + OFF_TAU;

  // running encoded min/max of depth endpoints seen by this wave (depths sorted along S)
  unsigned dlo = 0xFFFFFFFFu, dhi = 0x00000000u;

  int r = wid;
  if (r < NRAYS) stage_ray(colors, coords, dens, deps, r, l0, lane);

  int cur = 0;
  for (; r < NRAYS; r += stride, cur ^= 1) {
    int nxt = r + stride;
    if (nxt < NRAYS) {
      stage_ray(colors, coords, dens, deps, nxt, cur ? l0 : l1, lane);
      asm volatile("s_wait_asynccnt 24" ::: "memory");  // current buffer done, prefetch in flight
    } else {
      asm volatile("s_wait_asynccnt 0" ::: "memory");
    }
    const unsigned char* buf = cur ? buf1 : buf0;
    const float* C = (const float*)(buf + COL_OFF);
    const float* X = (const float*)(buf + CRD_OFF);
    const float* D = (const float*)(buf + DEN_OFF);
    const float* Z = (const float*)(buf + DEP_OFF);

    // global depth min/max tracking from staged data (broadcast LDS reads, zero HBM cost)
    {
      unsigned e0 = encf(Z[0]);
      unsigned e1 = encf(Z[SAMPLES - 1]);
      dlo = e0 < dlo ? e0 : dlo;
      dhi = e1 > dhi ? e1 : dhi;
    }

    float a[3], f[3], w[3], zm[3], cm[3][3], xm[3][3];
#pragma unroll
    for (int j = 0; j < 3; ++j) {
      int  s     = lane + 32 * j;
      bool valid = s < NMID;
      int  s1    = valid ? s + 1 : s;
      float z0 = Z[s], z1 = Z[s1];
      float d0 = D[s], d1 = D[s1];
      zm[j] = 0.5f * (z0 + z1);
      float dm = 0.5f * (d0 + d1);
      float sp = fmaxf(dm, 0.0f) + log1pf(expf(-fabsf(dm)));   // softplus, stable
      float e  = expf(-sp * (z1 - z0));
      a[j] = valid ? (1.0f - e) : 0.0f;
      f[j] = valid ? (e + 1e-10f) : 1.0f;
#pragma unroll
      for (int c = 0; c < 3; ++c) {
        cm[j][c] = 0.5f * (C[3 * s + c] + C[3 * s1 + c]);
        xm[j][c] = 0.5f * (X[3 * s + c] + X[3 * s1 + c]);
      }
    }

    // exclusive multiplicative scan over mids m = lane + 32*j (slot-major order)
    float incl[3], excl[3];
#pragma unroll
    for (int j = 0; j < 3; ++j) {
      float p = f[j];
#pragma unroll
      for (int d = 1; d < 32; d <<= 1) {
        float t = __shfl_up(p, d, 32);
        if (lane >= d) p *= t;
      }
      incl[j] = p;
      float e = __shfl_up(p, 1, 32);
      excl[j] = (lane == 0) ? 1.0f : e;
    }
    float t0 = __shfl(incl[0], 31, 32);
    float t1 = __shfl(incl[1], 31, 32);
    float tr0 = excl[0];
    float tr1 = t0 * excl[1];
    float tr2 = t0 * t1 * excl[2];
    w[0] = a[0] * tr0; w[1] = a[1] * tr1; w[2] = a[2] * tr2;
    float tau = t0 * t1 * __shfl(excl[2], 30, 32);  // trans[94] = prod_{k<94} f_k

    float rgb[3], pnt[3];
#pragma unroll
    for (int c = 0; c < 3; ++c) {
      rgb[c] = wave_sum(w[0] * cm[0][c] + w[1] * cm[1][c] + w[2] * cm[2][c]);
      pnt[c] = wave_sum(w[0] * xm[0][c] + w[1] * xm[1][c] + w[2] * xm[2][c]);
    }
    float depth = wave_sum(w[0] * zm[0] + w[1] * zm[1] + w[2] * zm[2]);
    float wtot  = wave_sum(w[0] + w[1] + w[2]);

    size_t wbase = (size_t)r * NMID;
#pragma unroll
    for (int j = 0; j < 3; ++j) {
      int m = lane + 32 * j;
      if (m < NMID) __builtin_nontemporal_store(w[j], &outW[wbase + m]);
    }
    if (lane == 0) {
      if (depth != depth) depth = __builtin_inff();   // nan_to_num(nan=inf); clamp in rm_clamp
#pragma unroll
      for (int c = 0; c < 3; ++c) {
        float rv = rgb[c];
        if (wb != 0) rv = rv + 1.0f - wtot;
        outRGB[3 * (size_t)r + c] = rv;
        outP[3 * (size_t)r + c]   = pnt[c];
      }
      outD[r] = depth;
      outT[r] = tau;
    }
  }

  // one atomic pair per wave: contribute this wave's depth range to the global range
  if (lane == 0) {
    atomicMin(ws + 0, dlo);
    atomicMax(ws + 1, dhi);
  }
}

extern "C" void kernel_launch(void* const* d_in, const int* in_sizes, int n_in,
                              void* d_out, int out_size, void* d_ws, size_t ws_size,
                              hipStream_t stream) {
  (void)in_sizes; (void)n_in; (void)out_size; (void)ws_size;
  const float* colors = (const float*)d_in[0];
  const float* dens   = (const float*)d_in[1];
  const float* deps   = (const float*)d_in[2];
  const float* coords = (const float*)d_in[3];
  const int*   wbp    = (const int*)d_in[4];
  float*    out = (float*)d_out;
  unsigned* ws  = (unsigned*)d_ws;

  rm_init_ws<<<1, 1, 0, stream>>>(ws);
  rm_main<<<BLOCKS, THREADS, 0, stream>>>(colors, dens, deps, coords, wbp, out, ws);
  rm_clamp<<<NRAYS / 256, 256, 0, stream>>>(out + OFF_DEPTH, ws);
}